// ContrastiveLoss_55637006352665
// MI455X (gfx1250) — compile-verified
//
#include <hip/hip_runtime.h>

// ---------------------------------------------------------------------------
// Supervised contrastive loss, n=8192, d=256, T=0.5.
// loss = (1/2n) * sum_{i!=j, label_i==label_j} [ log(exp(2*s_ij)+D_i) - 2*s_ij ]
//   s_ij = <x_i,x_j> * invn_i * invn_j,  D_i = sum_{label_j!=label_i} exp(2*s_ij)
// Two WMMA-f32 GEMM passes over X*X^T (34 GFLOP each, 8 MB working set -> L2
// resident); the 67M-element exp/log epilogue dominates, so loads must be
// hidden: each wave computes a 16x64 strip (A frag reused over 4 independent
// WMMA accumulators) and the epilogue uses fast __expf/__logf (args are in
// [-2,2] / [e^-2,5e4] so the HW trans ops are accurate). All reductions are
// single-writer or fixed-order => deterministic.
// ---------------------------------------------------------------------------

typedef float v2f __attribute__((ext_vector_type(2)));
typedef float v8f __attribute__((ext_vector_type(8)));

#define DIM        256
#define TEMP_INV   2.0f      // 1 / TEMPERATURE
#define EPS_NORM   1e-8f
#define MB         128       // rows per block (8 waves x 16)
#define CHUNK_COLS 512       // columns per block (8 groups of 4 j-tiles)
#define NCHUNK     16        // 8192 / CHUNK_COLS

// ---- 1 / max(||x_i||, eps), one wave per row ------------------------------
__global__ __launch_bounds__(256) void cl_invnorm(const float* __restrict__ X,
                                                  float* __restrict__ invn, int n) {
  int wave = threadIdx.x >> 5, lane = threadIdx.x & 31;
  int row = blockIdx.x * 8 + wave;
  if (row >= n) return;
  const float* xr = X + (size_t)row * DIM;
  float s = 0.f;
#pragma unroll
  for (int k = lane; k < DIM; k += 32) { float v = xr[k]; s += v * v; }
#pragma unroll
  for (int m = 16; m >= 1; m >>= 1) s += __shfl_xor(s, m, 32);
  if (lane == 0) invn[row] = 1.0f / fmaxf(sqrtf(s), EPS_NORM);
}

// ---- Pass 1: partial D_i per 512-col chunk (single writer per entry) ------
__global__ __launch_bounds__(256) void cl_pass1(const float* __restrict__ X,
                                                const float* __restrict__ invn,
                                                const int* __restrict__ label,
                                                float* __restrict__ Dpart, int n) {
  const int wave = threadIdx.x >> 5, lane = threadIdx.x & 31;
  const int half = lane >> 4, l16 = lane & 15;
  const int m0 = blockIdx.x * MB + wave * 16;
  const int jbase = blockIdx.y * CHUNK_COLS;

  float invn2_i[8]; int lab_i[8];
#pragma unroll
  for (int r = 0; r < 8; ++r) {          // C-frag row r lives at i = m0+8*half+r
    int i = m0 + 8 * half + r;
    invn2_i[r] = invn[i] * TEMP_INV;     // fold 1/T into the row scale
    lab_i[r] = label[i];
  }
  float dacc[8];
#pragma unroll
  for (int r = 0; r < 8; ++r) dacc[r] = 0.f;

  const float* arow = X + (size_t)(m0 + l16) * DIM + 2 * half;  // A frag base

  for (int g = 0; g < CHUNK_COLS / 64; ++g) {
    const int j0 = jbase + g * 64;
    const float* bp0 = X + (size_t)(j0 + l16) * DIM + 2 * half;
    const float* bp1 = bp0 + 16 * DIM;
    const float* bp2 = bp0 + 32 * DIM;
    const float* bp3 = bp0 + 48 * DIM;
    v8f cc0 = {}, cc1 = {}, cc2 = {}, cc3 = {};
#pragma unroll 4
    for (int ks = 0; ks < DIM / 4; ++ks) {
      const int o = 4 * ks;
      v2f a  = *(const v2f*)(arow + o);   // 1 A load feeds 4 independent WMMAs
      v2f b0 = *(const v2f*)(bp0 + o);
      v2f b1 = *(const v2f*)(bp1 + o);
      v2f b2 = *(const v2f*)(bp2 + o);
      v2f b3 = *(const v2f*)(bp3 + o);
      cc0 = __builtin_amdgcn_wmma_f32_16x16x4_f32(false, a, false, b0, (short)0, cc0, false, false);
      cc1 = __builtin_amdgcn_wmma_f32_16x16x4_f32(false, a, false, b1, (short)0, cc1, false, false);
      cc2 = __builtin_amdgcn_wmma_f32_16x16x4_f32(false, a, false, b2, (short)0, cc2, false, false);
      cc3 = __builtin_amdgcn_wmma_f32_16x16x4_f32(false, a, false, b3, (short)0, cc3, false, false);
    }
    v8f cc[4] = {cc0, cc1, cc2, cc3};
#pragma unroll
    for (int t = 0; t < 4; ++t) {
      const int j = j0 + 16 * t + l16;
      const float invn_j = invn[j];
      const int lab_j = label[j];
#pragma unroll
      for (int r = 0; r < 8; ++r) {
        float s2 = cc[t][r] * invn2_i[r] * invn_j;
        float e = __expf(s2);                      // s2 in [-2,2]: fast path safe
        dacc[r] += (lab_i[r] != lab_j) ? e : 0.f;  // diagonal is same-label: excluded
      }
    }
  }
  // lanes {0..15} / {16..31} each share one row per r: reduce within the half
#pragma unroll
  for (int r = 0; r < 8; ++r) {
    float v = dacc[r];
    v += __shfl_xor(v, 1, 32);
    v += __shfl_xor(v, 2, 32);
    v += __shfl_xor(v, 4, 32);
    v += __shfl_xor(v, 8, 32);
    if (l16 == 0) Dpart[(size_t)blockIdx.y * n + (m0 + 8 * half + r)] = v;
  }
}

// ---- Fixed-order chunk reduction: D[i] = sum_c Dpart[c][i] ----------------
__global__ __launch_bounds__(256) void cl_dreduce(const float* __restrict__ Dpart,
                                                  float* __restrict__ D, int n) {
  int i = blockIdx.x * 256 + threadIdx.x;
  if (i >= n) return;
  float s = 0.f;
#pragma unroll
  for (int c = 0; c < NCHUNK; ++c) s += Dpart[(size_t)c * n + i];
  D[i] = s;
}

// ---- Pass 2: per-wave loss partials (single writer per entry) -------------
__global__ __launch_bounds__(256) void cl_pass2(const float* __restrict__ X,
                                                const float* __restrict__ invn,
                                                const int* __restrict__ label,
                                                const float* __restrict__ D,
                                                float* __restrict__ losspart, int n) {
  const int wave = threadIdx.x >> 5, lane = threadIdx.x & 31;
  const int half = lane >> 4, l16 = lane & 15;
  const int m0 = blockIdx.x * MB + wave * 16;
  const int jbase = blockIdx.y * CHUNK_COLS;

  float invn2_i[8], Di[8]; int lab_i[8];
#pragma unroll
  for (int r = 0; r < 8; ++r) {
    int i = m0 + 8 * half + r;
    invn2_i[r] = invn[i] * TEMP_INV;
    lab_i[r] = label[i];
    Di[r] = D[i];
  }
  float lacc = 0.f;
  const float* arow = X + (size_t)(m0 + l16) * DIM + 2 * half;

  for (int g = 0; g < CHUNK_COLS / 64; ++g) {
    const int j0 = jbase + g * 64;
    const float* bp0 = X + (size_t)(j0 + l16) * DIM + 2 * half;
    const float* bp1 = bp0 + 16 * DIM;
    const float* bp2 = bp0 + 32 * DIM;
    const float* bp3 = bp0 + 48 * DIM;
    v8f cc0 = {}, cc1 = {}, cc2 = {}, cc3 = {};
#pragma unroll 4
    for (int ks = 0; ks < DIM / 4; ++ks) {
      const int o = 4 * ks;
      v2f a  = *(const v2f*)(arow + o);
      v2f b0 = *(const v2f*)(bp0 + o);
      v2f b1 = *(const v2f*)(bp1 + o);
      v2f b2 = *(const v2f*)(bp2 + o);
      v2f b3 = *(const v2f*)(bp3 + o);
      cc0 = __builtin_amdgcn_wmma_f32_16x16x4_f32(false, a, false, b0, (short)0, cc0, false, false);
      cc1 = __builtin_amdgcn_wmma_f32_16x16x4_f32(false, a, false, b1, (short)0, cc1, false, false);
      cc2 = __builtin_amdgcn_wmma_f32_16x16x4_f32(false, a, false, b2, (short)0, cc2, false, false);
      cc3 = __builtin_amdgcn_wmma_f32_16x16x4_f32(false, a, false, b3, (short)0, cc3, false, false);
    }
    v8f cc[4] = {cc0, cc1, cc2, cc3};
#pragma unroll
    for (int t = 0; t < 4; ++t) {
      const int j = j0 + 16 * t + l16;
      const float invn_j = invn[j];
      const int lab_j = label[j];
#pragma unroll
      for (int r = 0; r < 8; ++r) {
        const int i = m0 + 8 * half + r;
        float s2 = cc[t][r] * invn2_i[r] * invn_j;
        float v = __logf(__expf(s2) + Di[r]) - s2;          // -log(e/(e+D))
        lacc += (lab_i[r] == lab_j && i != j) ? v : 0.f;    // branchless select
      }
    }
  }
#pragma unroll
  for (int m = 16; m >= 1; m >>= 1) lacc += __shfl_xor(lacc, m, 32);
  if (lane == 0)
    losspart[((size_t)blockIdx.y * gridDim.x + blockIdx.x) * 8 + wave] = lacc;
}

// ---- Fixed-order final reduction + scale ----------------------------------
__global__ __launch_bounds__(256) void cl_finalize(const float* __restrict__ losspart,
                                                   float* __restrict__ out,
                                                   int nterms, float scale) {
  __shared__ float sm[256];
  float s = 0.f;
  for (int idx = threadIdx.x; idx < nterms; idx += 256) s += losspart[idx];
  sm[threadIdx.x] = s;
  __syncthreads();
  for (int step = 128; step >= 1; step >>= 1) {
    if (threadIdx.x < step) sm[threadIdx.x] += sm[threadIdx.x + step];
    __syncthreads();
  }
  if (threadIdx.x == 0) out[0] = sm[0] * scale;
}

extern "C" void kernel_launch(void* const* d_in, const int* in_sizes, int n_in,
                              void* d_out, int out_size, void* d_ws, size_t ws_size,
                              hipStream_t stream) {
  const float* X = (const float*)d_in[0];   // logits [n,256] f32
  const int* label = (const int*)d_in[1];   // labels [n] (harness delivers int32)
  float* out = (float*)d_out;               // scalar f32
  const int n = in_sizes[1];                // 8192

  // workspace: invn[n] | Dpart[NCHUNK*n] | D[n] | losspart[(n/MB)*NCHUNK*8]
  float* invn = (float*)d_ws;
  float* Dpart = invn + n;
  float* Dfull = Dpart + (size_t)NCHUNK * n;
  float* losspart = Dfull + n;
  const int nterms = (n / MB) * NCHUNK * 8;

  cl_invnorm<<<n / 8, 256, 0, stream>>>(X, invn, n);
  dim3 grid(n / MB, NCHUNK);
  cl_pass1<<<grid, 256, 0, stream>>>(X, invn, label, Dpart, n);
  cl_dreduce<<<(n + 255) / 256, 256, 0, stream>>>(Dpart, Dfull, n);
  cl_pass2<<<grid, 256, 0, stream>>>(X, invn, label, Dfull, losspart, n);
  cl_finalize<<<1, 256, 0, stream>>>(losspart, out, nterms, 1.0f / (2.0f * n));
}